// BGEM3Model_16054587752450
// MI455X (gfx1250) — compile-verified
//
#include <hip/hip_runtime.h>
#include <hip/hip_bf16.h>

typedef __attribute__((ext_vector_type(2))) float v2f;
typedef __attribute__((ext_vector_type(8))) float v8f;

#define BATCH 32
#define SEQ   1024
#define HID   1024
#define VOCAB 250002
#define EPSF  1e-12f

// d_out layout (flat f32, in reference return order)
#define DENSE_OFF   0
#define SPARSE_OFF  ((size_t)BATCH * HID)
#define COLBERT_OFF ((size_t)BATCH * HID + (size_t)BATCH * VOCAB)

// ---------------------------------------------------------------- reductions
__device__ __forceinline__ float wave_reduce_sum(float v) {
#pragma unroll
  for (int off = 16; off; off >>= 1) v += __shfl_xor(v, off, 32);
  return v;
}

__device__ __forceinline__ float block_reduce_sum_256(float v, float* smem8) {
  int lane = threadIdx.x & 31;
  int wid = threadIdx.x >> 5;
  v = wave_reduce_sum(v);
  if (lane == 0) smem8[wid] = v;
  __syncthreads();
  float t = 0.f;
#pragma unroll
  for (int i = 0; i < 8; ++i) t += smem8[i];
  __syncthreads();
  return t;
}

// ---------------------------------------------------------------- 1) zero sparse region
__global__ void zero_sparse_kernel(float* __restrict__ out) {
  size_t i = (size_t)blockIdx.x * blockDim.x + threadIdx.x;
  if (i < (size_t)BATCH * VOCAB) out[SPARSE_OFF + i] = 0.0f;
}

// ---------------------------------------------------------------- 2) dense: CLS + l2norm
__global__ __launch_bounds__(256) void dense_kernel(const float* __restrict__ lhs,
                                                    float* __restrict__ out) {
  __shared__ float smem[8];
  int b = blockIdx.x;
  int t = threadIdx.x;                       // 256 threads * float4 = 1024
  const float4* row = (const float4*)(lhs + (size_t)b * SEQ * HID);  // s = 0
  float4 v = row[t];
  float ss = v.x * v.x + v.y * v.y + v.z * v.z + v.w * v.w;
  float total = block_reduce_sum_256(ss, smem);
  float scale = 1.0f / fmaxf(sqrtf(total), EPSF);
  float4 o = make_float4(v.x * scale, v.y * scale, v.z * scale, v.w * scale);
  ((float4*)(out + DENSE_OFF + (size_t)b * HID))[t] = o;
}

// ---------------------------------------------------------------- 3) sparse: relu(lin) -> scatter-max
__global__ __launch_bounds__(256) void sparse_kernel(const float* __restrict__ lhs,
                                                     const int* __restrict__ ids,
                                                     const float* __restrict__ sW,
                                                     const float* __restrict__ sb,
                                                     float* __restrict__ out) {
  int wave = threadIdx.x >> 5;
  int lane = threadIdx.x & 31;
  int token = blockIdx.x * 8 + wave;         // one wave per token
  int b = token >> 10;
  int s = token & 1023;
  const float4* row = (const float4*)(lhs + ((size_t)b * SEQ + s) * HID);
  const float4* w4 = (const float4*)sW;
  float acc = 0.f;
#pragma unroll
  for (int j = 0; j < 8; ++j) {              // 256 float4 / 32 lanes
    float4 x = row[lane + 32 * j];
    float4 w = w4[lane + 32 * j];
    acc += x.x * w.x + x.y * w.y + x.z * w.z + x.w * w.w;
  }
  acc = wave_reduce_sum(acc);
  if (lane == 0) {
    float tw = acc + sb[0];
    if (tw > 0.0f) {
      int id = ids[(size_t)b * SEQ + s];
      if (id >= 4) {                         // unused tokens {0,1,2,3} stay 0
        // tw >= 0 and buffer zero-inited: u32 max == f32 max
        atomicMax((unsigned int*)(out + SPARSE_OFF + (size_t)b * VOCAB + id),
                  __float_as_uint(tw));
      }
    }
  }
}

// ---------------------------------------------------------------- 4) colbert GEMM (WMMA f32)
// C[b][r][o] = sum_h lhs[b][1+r][h] * W[o][h],  r in [0,1023), raw (no bias/mask/norm)
#define BM 64
#define BN 64
#define BK 32
#define LDP 36   // pitch: multiple of 4 (16B-aligned float4 stores), conflict-free frag reads

__global__ __launch_bounds__(256) void colbert_gemm_kernel(const float* __restrict__ lhs,
                                                           const float* __restrict__ W,
                                                           float* __restrict__ out) {
  __shared__ float As[BM][LDP];
  __shared__ float Ws[BN][LDP];

  int b  = blockIdx.z;
  int m0 = blockIdx.y * BM;                  // row-tile (r index)
  int n0 = blockIdx.x * BN;                  // col-tile (o index)

  int tid  = threadIdx.x;
  int lane = tid & 31;
  int wave = tid >> 5;
  int wr = wave >> 1;                        // 0..3  (16-row strip)
  int wc = wave & 1;                         // 0..1  (32-col strip)
  int lm = lane & 15;                        // M (A) / N (B) within tile
  int kh = lane >> 4;                        // K-half: 0 -> K{0,1}, 1 -> K{2,3}

  v8f acc0 = {0.f, 0.f, 0.f, 0.f, 0.f, 0.f, 0.f, 0.f};
  v8f acc1 = {0.f, 0.f, 0.f, 0.f, 0.f, 0.f, 0.f, 0.f};

  int lrow = tid >> 3;                       // 0..31
  int lcol = (tid & 7) * 4;                  // 0,4,...,28

  const float* Abase = lhs + (size_t)b * SEQ * HID;

  for (int k0 = 0; k0 < HID; k0 += BK) {
#pragma unroll
    for (int p = 0; p < 2; ++p) {            // 64 rows in two passes of 32
      int row = p * 32 + lrow;
      // Clamp instead of predicate: row r=1023 (only in the last M-tile) reads
      // the valid row 1022; its C outputs are never stored. Keeps the inner
      // pipeline free of EXEC-mask branches.
      int r = m0 + row;
      int rs = (r < 1023) ? r : 1022;
      float4 av = *(const float4*)(Abase + (size_t)(1 + rs) * HID + k0 + lcol);
      *(float4*)&As[row][lcol] = av;
      int n = n0 + row;
      float4 wv = *(const float4*)(W + (size_t)n * HID + k0 + lcol);
      *(float4*)&Ws[row][lcol] = wv;
    }
    __syncthreads();
#pragma unroll
    for (int kk = 0; kk < BK; kk += 4) {
      v2f a  = *(const v2f*)&As[wr * 16 + lm][kk + 2 * kh];
      v2f b0 = *(const v2f*)&Ws[wc * 32 + lm][kk + 2 * kh];
      v2f b1 = *(const v2f*)&Ws[wc * 32 + 16 + lm][kk + 2 * kh];
      acc0 = __builtin_amdgcn_wmma_f32_16x16x4_f32(false, a, false, b0,
                                                   (short)0, acc0, false, false);
      acc1 = __builtin_amdgcn_wmma_f32_16x16x4_f32(false, a, false, b1,
                                                   (short)0, acc1, false, false);
    }
    __syncthreads();
  }

  // C layout: VGPR i -> lanes 0-15: (M=i, N=lane); lanes 16-31: (M=i+8, N=lane-16)
  float* outc = out + COLBERT_OFF;
  int c0 = n0 + wc * 32 + lm;
#pragma unroll
  for (int i = 0; i < 8; ++i) {
    int r = m0 + wr * 16 + i + 8 * kh;
    if (r < 1023) {
      size_t base = ((size_t)b * 1023 + r) * HID;
      outc[base + c0] = acc0[i];
      outc[base + c0 + 16] = acc1[i];
    }
  }
}

// ---------------------------------------------------------------- 5) colbert finalize: +bias, *mask, l2norm (in place)
__global__ __launch_bounds__(256) void colbert_finalize_kernel(const float* __restrict__ bias,
                                                               const int* __restrict__ mask,
                                                               float* __restrict__ out) {
  __shared__ float smem[8];
  int rowid = blockIdx.x;                    // 0 .. 32*1023-1
  int b = rowid / 1023;
  int r = rowid % 1023;
  int t = threadIdx.x;
  float* row = out + COLBERT_OFF + (size_t)rowid * HID;
  float m = (float)mask[(size_t)b * SEQ + (1 + r)];
  float4 v = ((const float4*)row)[t];
  float4 bb = ((const float4*)bias)[t];
  v.x = (v.x + bb.x) * m;
  v.y = (v.y + bb.y) * m;
  v.z = (v.z + bb.z) * m;
  v.w = (v.w + bb.w) * m;
  float ss = v.x * v.x + v.y * v.y + v.z * v.z + v.w * v.w;
  float total = block_reduce_sum_256(ss, smem);
  float scale = 1.0f / fmaxf(sqrtf(total), EPSF);
  v.x *= scale; v.y *= scale; v.z *= scale; v.w *= scale;
  ((float4*)row)[t] = v;
}

// ---------------------------------------------------------------- launch
extern "C" void kernel_launch(void* const* d_in, const int* in_sizes, int n_in,
                              void* d_out, int out_size, void* d_ws, size_t ws_size,
                              hipStream_t stream) {
  (void)in_sizes; (void)n_in; (void)out_size; (void)d_ws; (void)ws_size;
  const float* lhs       = (const float*)d_in[0];   // [32,1024,1024] f32
  const int*   attn_mask = (const int*)d_in[1];     // [32,1024] i32
  const int*   input_ids = (const int*)d_in[2];     // [32,1024] i32
  const float* colbert_W = (const float*)d_in[3];   // [1024,1024] f32 (o-major)
  const float* colbert_b = (const float*)d_in[4];   // [1024] f32
  const float* sparse_W  = (const float*)d_in[5];   // [1024] f32
  const float* sparse_b  = (const float*)d_in[6];   // [1] f32
  float* out = (float*)d_out;

  // 1) zero sparse output region
  {
    size_t n = (size_t)BATCH * VOCAB;
    int blocks = (int)((n + 255) / 256);
    zero_sparse_kernel<<<blocks, 256, 0, stream>>>(out);
  }
  // 2) dense head
  dense_kernel<<<BATCH, 256, 0, stream>>>(lhs, out);
  // 3) sparse head (after zero init, same stream -> ordered)
  sparse_kernel<<<(BATCH * SEQ) / 8, 256, 0, stream>>>(lhs, input_ids, sparse_W,
                                                       sparse_b, out);
  // 4) colbert GEMM: grid (Ntiles, Mtiles, batch)
  {
    dim3 grid(HID / BN, (1023 + BM - 1) / BM, BATCH);
    colbert_gemm_kernel<<<grid, 256, 0, stream>>>(lhs, colbert_W, out);
  }
  // 5) colbert finalize
  colbert_finalize_kernel<<<BATCH * 1023, 256, 0, stream>>>(colbert_b, attn_mask, out);
}